// Iteration_18408229830968
// MI455X (gfx1250) — compile-verified
//
#include <hip/hip_runtime.h>

// MI455X / gfx1250 implementation.
// 3x3 circular conv executed on the WMMA pipe as banded f32 matmuls
// (V_WMMA_F32_16X16X4_F32); halo tiles staged global->LDS with async-LDS
// loads (ASYNCcnt) when the toolchain exposes them; epilogues fused on the
// VALU with prefetched operands. Everything fp32, L2-resident
// (192MB L2 >> ~112MB working set).

#define HH 2048
#define WW 2048
#define YMASK 2047
#define XMASK 2047

typedef float v2f __attribute__((ext_vector_type(2)));
typedef float v8f __attribute__((ext_vector_type(8)));

#define LSTRIDE 68   // 66 data cols + 2 zero cols (K-tail); stride 68 -> 4-bank step, conflict free
#define LROWS   66
#define LCOLS   66

#if __has_builtin(__builtin_amdgcn_global_load_async_to_lds_b32) && \
    __has_builtin(__builtin_amdgcn_s_wait_asynccnt)
#define USE_ASYNC_LDS 1
#else
#define USE_ASYNC_LDS 0
#endif

typedef __attribute__((address_space(1))) int* gptr_t;
typedef __attribute__((address_space(3))) int* lptr_t;

// ---------------------------------------------------------------------------
// Fused conv kernel. One 64x64 output region per 256-thread block (8 waves,
// 2 WMMA tiles of 16x16 per wave).
//   MODE 0: base_new = sigma_expr + alpha*conv(src=STg, flip(h));  acc = g0*base_new
//   MODE 1: tdst     = decimate(conv(src=base, h))
//   MODE 2: base_new = alpha*conv(src=t, flip(h)) + (beta0+sigma)*lap(base_old);
//           acc     += g_i*base_new
// ---------------------------------------------------------------------------
template <int MODE, int FLIP>
__global__ __launch_bounds__(256)
void conv3x3_fused_kernel(const float* __restrict__ src,
                          const float* __restrict__ wgt,
                          float* __restrict__ tdst,
                          const float* __restrict__ base_old,
                          float* __restrict__ base_new,
                          float* __restrict__ acc,
                          const float* __restrict__ d_x, const float* __restrict__ d_y,
                          const float* __restrict__ b_x, const float* __restrict__ b_y,
                          const float* __restrict__ alpha_p, const float* __restrict__ beta0_p,
                          const float* __restrict__ sigma_p, const float* __restrict__ gamma_p,
                          int iter,
                          const int* __restrict__ dr_p, const int* __restrict__ dc_p)
{
    __shared__ float smem[LROWS * LSTRIDE];

    const int tid  = threadIdx.x;
    const int lane = tid & 31;
    const int wave = tid >> 5;
    const int y0   = blockIdx.y * 64;
    const int x0   = blockIdx.x * 64;

    // uniform scalar parameters (device-resident, graph-capture safe)
    const float alpha = alpha_p[0];
    const float beta0 = beta0_p[0];
    const float sigma = sigma_p[0];
    const float gcoef = gamma_p[iter];
    const int   dr    = (MODE == 1) ? dr_p[0] : 1;
    const int   dc    = (MODE == 1) ? dc_p[0] : 1;

    // ---- prefetch epilogue operands (overlaps with staging + WMMAs) --------
    if (tid < 128) {
        const int ry = (y0 + (tid >> 1)) & YMASK;
        const int rx = x0 + (tid & 1) * 32;          // 128B cachelines
        const size_t o = (size_t)ry * WW + rx;
        if (MODE == 0) {
            __builtin_prefetch(&d_x[o], 0, 3);
            __builtin_prefetch(&d_y[o], 0, 3);
            __builtin_prefetch(&b_x[o], 0, 3);
            __builtin_prefetch(&b_y[o], 0, 3);
        } else if (MODE == 2) {
            __builtin_prefetch(&base_old[o], 0, 3);
        }
    }

    // ---- stage 66x66 wrapped halo tile into LDS; cols 66..67 zeroed --------
    for (int i = tid; i < LROWS * LSTRIDE; i += 256) {
        const int ly = i / LSTRIDE;
        const int lx = i - ly * LSTRIDE;
        if (lx < LCOLS) {
            const int gy = (y0 - 1 + ly) & YMASK;
            const int gx = (x0 - 1 + lx) & XMASK;
#if USE_ASYNC_LDS
            __builtin_amdgcn_global_load_async_to_lds_b32(
                (gptr_t)(src + (size_t)gy * WW + gx),
                (lptr_t)(&smem[i]),
                0, 0);
#else
            smem[i] = src[(size_t)gy * WW + gx];
#endif
        } else {
            smem[i] = 0.0f;
        }
    }
#if USE_ASYNC_LDS
    __builtin_amdgcn_s_wait_asynccnt(0);
#endif
    __syncthreads();

    // ---- weights: compile-time flip -> plain uniform loads into SGPRs ------
    float wrow[3][3];
#pragma unroll
    for (int i = 0; i < 3; ++i)
#pragma unroll
        for (int j = 0; j < 3; ++j)
            wrow[i][j] = FLIP ? wgt[(2 - i) * 3 + (2 - j)] : wgt[i * 3 + j];

    const int n    = lane & 15;   // output column within tile (B/C/D layout)
    const int half = lane >> 4;   // K-half selector (A/B layout)

    // ---- banded B matrices: M_dy[p][n] = wrow[dy][p-n], 0<=p-n<=2 ----------
    // B 4x16 layout: VGPR0 holds K-rows {0,2}, VGPR1 K-rows {1,3};
    // lanes 0-15 -> first row of the pair, lanes 16-31 -> second.
    // Masks depend only on (kb, lane): hoisted outside dy so they CSE.
    v2f bm[3][5];
#pragma unroll
    for (int kb = 0; kb < 5; ++kb) {
        const int d0 = kb * 4 + half * 2 - n;
        const int d1 = d0 + 1;
        const bool x0m = (d0 == 0), x1m = (d0 == 1), x2m = (d0 == 2);
        const bool y0m = (d1 == 0), y1m = (d1 == 1), y2m = (d1 == 2);
#pragma unroll
        for (int dy = 0; dy < 3; ++dy) {
            bm[dy][kb].x = x0m ? wrow[dy][0] : (x1m ? wrow[dy][1] : (x2m ? wrow[dy][2] : 0.0f));
            bm[dy][kb].y = y0m ? wrow[dy][0] : (y1m ? wrow[dy][1] : (y2m ? wrow[dy][2] : 0.0f));
        }
    }

    // ---- two 16x16 tiles per wave; 15 chained f32 WMMAs per tile -----------
#pragma unroll
    for (int t = 0; t < 2; ++t) {
        const int tileIdx = wave * 2 + t;
        const int ty = tileIdx >> 2;
        const int tx = tileIdx & 3;

        v8f c = {0.f, 0.f, 0.f, 0.f, 0.f, 0.f, 0.f, 0.f};

#pragma unroll
        for (int dy = 0; dy < 3; ++dy) {
            // A 16x4 layout: lane m = lane&15 holds row m; half-wave selects K pair
            const int row = ty * 16 + n + dy;          // smem row 0 == global y0-1
            const float* rp = &smem[row * LSTRIDE + tx * 16];
#pragma unroll
            for (int kb = 0; kb < 5; ++kb) {
                const int p = kb * 4 + half * 2;
                v2f a;
                a.x = rp[p];
                a.y = rp[p + 1];
                c = __builtin_amdgcn_wmma_f32_16x16x4_f32(
                        /*neg_a=*/false, a, /*neg_b=*/false, bm[dy][kb],
                        /*c_mod=*/(short)0, c, /*reuse_a=*/false, /*reuse_b=*/false);
            }
        }

        // ---- epilogue: C/D layout -> VGPR r holds rows r (lanes 0-15) / r+8
#pragma unroll
        for (int r = 0; r < 8; ++r) {
            const int m   = (lane < 16) ? r : (r + 8);
            const int gy  = y0 + ty * 16 + m;
            const int gx  = x0 + tx * 16 + n;
            const size_t idx = (size_t)gy * WW + gx;
            const float v = c[r];

            if (MODE == 0) {
                const int xm = (gx - 1) & XMASK;
                const int ym = (gy - 1) & YMASK;
                const size_t ixm = (size_t)gy * WW + xm;
                const size_t iym = (size_t)ym * WW + gx;
                // sigma * (dxT(d_x-b_x) + dyT(d_y-b_y)):  v[x-1]-v[x] form
                const float dxT = (d_x[ixm] - b_x[ixm]) - (d_x[idx] - b_x[idx]);
                const float dyT = (d_y[iym] - b_y[iym]) - (d_y[idx] - b_y[idx]);
                const float xv  = sigma * (dxT + dyT) + alpha * v;
                base_new[idx] = xv;
                acc[idx]      = gcoef * xv;       // gamma[0] * base
            } else if (MODE == 1) {
                tdst[idx] = ((gy % dr) == 0 && (gx % dc) == 0) ? v : 0.0f;
            } else {
                const int xm = (gx - 1) & XMASK, xp = (gx + 1) & XMASK;
                const int ym = (gy - 1) & YMASK, yp = (gy + 1) & YMASK;
                const float u   = base_old[idx];
                // dxT(dx(u)) + dyT(dy(u)) = 4u - l - r - up - dn
                const float lap = 4.0f * u
                                - base_old[(size_t)gy * WW + xm]
                                - base_old[(size_t)gy * WW + xp]
                                - base_old[(size_t)ym * WW + gx]
                                - base_old[(size_t)yp * WW + gx];
                const float nv = alpha * v + (beta0 + sigma) * lap;
                base_new[idx] = nv;
                acc[idx]     += gcoef * nv;
            }
        }
    }
}

// ---------------------------------------------------------------------------
// Final shrinkage: f = acc; emit (f, d_x_new, d_y_new, b_x_new, b_y_new)
// ---------------------------------------------------------------------------
__global__ __launch_bounds__(256)
void final_shrink_kernel(const float* __restrict__ acc,
                         const float* __restrict__ b_x, const float* __restrict__ b_y,
                         const float* __restrict__ beta1_p, const float* __restrict__ sigma_p,
                         float* __restrict__ out)
{
    const size_t N   = (size_t)HH * WW;
    const size_t idx = (size_t)blockIdx.x * 256 + threadIdx.x;
    const int gy = (int)(idx >> 11);
    const int gx = (int)(idx & XMASK);

    const float thresh = beta1_p[0] / sigma_p[0];

    const float f  = acc[idx];
    const float fr = acc[(size_t)gy * WW + ((gx + 1) & XMASK)];
    const float fd = acc[(size_t)((gy + 1) & YMASK) * WW + gx];
    const float dxf = fr - f;
    const float dyf = fd - f;

    const float bx = b_x[idx];
    const float by = b_y[idx];
    const float px = dxf + bx;
    const float py = dyf + by;
    const float norm   = sqrtf(px * px + py * py);
    const float shrink = fmaxf(norm - thresh, 0.0f) / (norm + 1e-8f);
    const float dxn = px * shrink;
    const float dyn = py * shrink;

    out[idx]         = f;
    out[N + idx]     = dxn;
    out[2 * N + idx] = dyn;
    out[3 * N + idx] = bx + dxf - dxn;
    out[4 * N + idx] = by + dyf - dyn;
}

// ---------------------------------------------------------------------------
extern "C" void kernel_launch(void* const* d_in, const int* in_sizes, int n_in,
                              void* d_out, int out_size, void* d_ws, size_t ws_size,
                              hipStream_t stream)
{
    (void)in_sizes; (void)n_in; (void)out_size; (void)ws_size;

    const float* STg   = (const float*)d_in[0];
    const float* d_x   = (const float*)d_in[1];
    const float* d_y   = (const float*)d_in[2];
    const float* b_x   = (const float*)d_in[3];
    const float* b_y   = (const float*)d_in[4];
    const int*   drp   = (const int*)  d_in[5];
    const int*   dcp   = (const int*)  d_in[6];
    const float* alpha = (const float*)d_in[7];
    const float* beta0 = (const float*)d_in[8];
    const float* beta1 = (const float*)d_in[9];
    const float* sigma = (const float*)d_in[10];
    const float* gamma = (const float*)d_in[11];
    const float* hwt   = (const float*)d_in[12];
    float* out = (float*)d_out;

    const size_t N = (size_t)HH * WW;
    float* ws    = (float*)d_ws;
    float* acc   = ws;
    float* baseA = ws + N;
    float* baseB = ws + 2 * N;
    float* tbuf  = ws + 3 * N;

    dim3 cgrid(WW / 64, HH / 64);
    dim3 cblk(256);

    // x = sigma_expr + alpha*conv(STg, flip(h)); base=x; acc=gamma[0]*x
    conv3x3_fused_kernel<0, 1><<<cgrid, cblk, 0, stream>>>(
        STg, hwt, tbuf, baseA, baseA, acc,
        d_x, d_y, b_x, b_y, alpha, beta0, sigma, gamma, /*iter=*/0, drp, dcp);

    float* cur = baseA;
    float* nxt = baseB;
    for (int i = 1; i <= 5; ++i) {
        // t = decimate(conv(base, h))
        conv3x3_fused_kernel<1, 0><<<cgrid, cblk, 0, stream>>>(
            cur, hwt, tbuf, cur, nxt, acc,
            d_x, d_y, b_x, b_y, alpha, beta0, sigma, gamma, i, drp, dcp);
        // base' = alpha*conv(t, flip(h)) + (beta0+sigma)*lap(base); acc += gamma[i]*base'
        conv3x3_fused_kernel<2, 1><<<cgrid, cblk, 0, stream>>>(
            tbuf, hwt, tbuf, cur, nxt, acc,
            d_x, d_y, b_x, b_y, alpha, beta0, sigma, gamma, i, drp, dcp);
        float* tmp = cur; cur = nxt; nxt = tmp;
    }

    final_shrink_kernel<<<dim3((unsigned)(N / 256)), cblk, 0, stream>>>(
        acc, b_x, b_y, beta1, sigma, out);
}